// FastJacobianMetricLayer_73040213836080
// MI455X (gfx1250) — compile-verified
//
#include <hip/hip_runtime.h>
#include <math.h>

// Problem constants (match reference)
#define NN   2048
#define GG   16384
#define DD   64
#define KK   4
#define REG  1e-6f
#define EPSW 1e-6f
#define KEXT (DD * DD + DD)   // 4160: flattened Gi + (-2*Gz) extension

typedef __attribute__((ext_vector_type(2))) float v2f;
typedef __attribute__((ext_vector_type(8))) float v8f;

// D = A(16x4) * B(4x16) + C, fp32 WMMA
__device__ __forceinline__ v8f wmma_f32(v2f a, v2f b, v8f c) {
  return __builtin_amdgcn_wmma_f32_16x16x4_f32(false, a, false, b, (short)0, c,
                                               false, false);
}

// Async global->LDS copy of 16 bytes (CDNA5 GLOBAL_LOAD_ASYNC_TO_LDS_B128,
// tracked by ASYNCcnt; bypasses VGPR round-trip).
__device__ __forceinline__ void async_copy_b128(void* lds_dst,
                                                const void* gsrc) {
  unsigned loff = (unsigned)(unsigned long long)lds_dst;  // LDS aperture low32
  asm volatile("global_load_async_to_lds_b128 %0, %1, off"
               :
               : "v"(loff), "v"(gsrc)
               : "memory");
}
__device__ __forceinline__ void async_wait0() {
  asm volatile("s_wait_asynccnt 0x0" ::: "memory");
}

// ---------------------------------------------------------------------------
// Kernel 0: g_sq[g] = ||grid[g]||^2
// ---------------------------------------------------------------------------
__global__ __launch_bounds__(256) void gsq_kernel(
    const float* __restrict__ grid, float* __restrict__ gsq) {
  int g = blockIdx.x * 256 + threadIdx.x;
  if (g < GG) {
    const float* p = grid + (size_t)g * DD;
    float s = 0.f;
#pragma unroll
    for (int d = 0; d < DD; ++d) s += p[d] * p[d];
    gsq[g] = s;
  }
}

// ---------------------------------------------------------------------------
// Kernel 1: kNN.  One block per 16 rows of z.  WMMA computes dot(z, grid^T)
// tile-by-tile; 16 scan threads keep running top-4 of (g_sq - 2*dot).
// ---------------------------------------------------------------------------
__global__ __launch_bounds__(256) void knn_kernel(
    const float* __restrict__ z, const float* __restrict__ grid,
    const float* __restrict__ gsq, int* __restrict__ nbr_idx,
    float* __restrict__ nbr_dist) {
  __shared__ float zt[16][DD];     // 16 z rows
  __shared__ float d2s[16][132];   // adjusted-d2 tile (128 cols + pad)
  __shared__ float gss[128];

  const int tid  = threadIdx.x;
  const int lane = tid & 31;
  const int wave = tid >> 5;
  const int i0   = blockIdx.x * 16;
  const int m    = lane & 15;
  const int kh   = (lane >> 4) << 1;  // 0 or 2 (K half per ISA A/B layout)
  const int chi  = lane >> 4;

  for (int p = tid; p < 16 * DD; p += 256)
    zt[p >> 6][p & 63] = z[(size_t)(i0 + (p >> 6)) * DD + (p & 63)];
  __syncthreads();

  float best[KK]  = {3.4e38f, 3.4e38f, 3.4e38f, 3.4e38f};
  int   besti[KK] = {0, 0, 0, 0};
  float zsq = 0.f;
  if (tid < 16) {
#pragma unroll
    for (int d = 0; d < DD; ++d) zsq += zt[tid][d] * zt[tid][d];
  }

  for (int it = 0; it < GG / 128; ++it) {
    const int g0 = it * 128;
    if (tid < 128) gss[tid] = gsq[g0 + tid];

    // 16x16 dot tile per wave, K = 64 in steps of 4
    v8f acc = {};
    const float* gb = grid + (size_t)(g0 + wave * 16 + m) * DD;
#pragma unroll
    for (int kk = 0; kk < DD; kk += 4) {
      v2f a, b;
      a.x = zt[m][kk + kh];
      a.y = zt[m][kk + kh + 1];
      b.x = gb[kk + kh];
      b.y = gb[kk + kh + 1];
      acc = wmma_f32(a, b, acc);
    }
    __syncthreads();  // gss visible; previous scan finished (d2s reusable)
#pragma unroll
    for (int v = 0; v < 8; ++v) {
      int row = v + 8 * chi;  // C layout: vgpr v -> M=v (lo lanes) / v+8 (hi)
      d2s[row][wave * 16 + m] = gss[wave * 16 + m] - 2.0f * acc[v];
    }
    __syncthreads();
    if (tid < 16) {
      for (int c = 0; c < 128; ++c) {
        float vv = d2s[tid][c];
        if (vv < best[KK - 1]) {
          best[KK - 1]  = vv;
          besti[KK - 1] = g0 + c;
#pragma unroll
          for (int q = KK - 1; q > 0; --q) {
            if (best[q] < best[q - 1]) {
              float tv = best[q]; best[q] = best[q - 1]; best[q - 1] = tv;
              int   ti = besti[q]; besti[q] = besti[q - 1]; besti[q - 1] = ti;
            }
          }
        }
      }
    }
    __syncthreads();
  }

  if (tid < 16) {
#pragma unroll
    for (int k = 0; k < KK; ++k) {
      nbr_idx[(i0 + tid) * KK + k] = besti[k];
      float d2 = zsq + best[k];
      nbr_dist[(i0 + tid) * KK + k] = sqrtf(fmaxf(d2, 0.f));
    }
  }
}

// ---------------------------------------------------------------------------
// Kernel 2: per node n:  Gi = sum_k w_k J_k^T J_k + REG*I  (WMMA),
// Gz = Gi z_n, a[n] = z_n . Gz.  Writes row n of Gi_ext[N, 4160]:
//   cols [0,4096)  = Gi flattened, cols [4096,4160) = -2*Gz.
// ---------------------------------------------------------------------------
__global__ __launch_bounds__(256) void gi_kernel(
    const float* __restrict__ z, const float* __restrict__ jac,
    const int* __restrict__ nbr_idx, const float* __restrict__ nbr_dist,
    float* __restrict__ gi_ext, float* __restrict__ a_arr) {
  __shared__ float Jt[64][65];  // staged Jacobian, then reused for Gi
  __shared__ float zn[64];
  __shared__ float gz[64];
  __shared__ float wsh[KK];
  __shared__ int   ish[KK];

  const int n    = blockIdx.x;
  const int tid  = threadIdx.x;
  const int lane = tid & 31;
  const int wave = tid >> 5;
  const int m    = lane & 15;
  const int kh   = (lane >> 4) << 1;
  const int chi  = lane >> 4;

  if (tid < 64) zn[tid] = z[(size_t)n * DD + tid];
  if (tid == 0) {
    float wv[KK], wsum = 0.f;
#pragma unroll
    for (int k = 0; k < KK; ++k) {
      float dk = nbr_dist[n * KK + k];
      wv[k] = 1.f / (dk + EPSW);
      wsum += wv[k];
      ish[k] = nbr_idx[n * KK + k];
    }
#pragma unroll
    for (int k = 0; k < KK; ++k) wsh[k] = wv[k] / wsum;
  }
  __syncthreads();

  // two 16x16 output tiles per wave (same ti, adjacent tj -> shared A frag)
  const int tile0 = wave * 2;
  const int ti  = tile0 >> 2;
  const int tj0 = tile0 & 3;
  const int tj1 = tj0 + 1;

  v8f acc0 = {}, acc1 = {};
  for (int k = 0; k < KK; ++k) {
    __syncthreads();  // Jt free
    const float* Jp = jac + (size_t)ish[k] * (DD * DD);
    for (int p = tid; p < DD * DD; p += 256) Jt[p >> 6][p & 63] = Jp[p];
    __syncthreads();
    const float wk = wsh[k];
#pragma unroll
    for (int kk = 0; kk < DD; kk += 4) {
      v2f a, b0, b1;
      a.x  = Jt[kk + kh][ti * 16 + m];
      a.y  = Jt[kk + kh + 1][ti * 16 + m];
      b0.x = Jt[kk + kh][tj0 * 16 + m] * wk;
      b0.y = Jt[kk + kh + 1][tj0 * 16 + m] * wk;
      b1.x = Jt[kk + kh][tj1 * 16 + m] * wk;
      b1.y = Jt[kk + kh + 1][tj1 * 16 + m] * wk;
      acc0 = wmma_f32(a, b0, acc0);
      acc1 = wmma_f32(a, b1, acc1);
    }
  }
  __syncthreads();  // everyone done reading Jt; reuse it for Gi

  float* grow = gi_ext + (size_t)n * KEXT;
#pragma unroll
  for (int v = 0; v < 8; ++v) {
    int r  = ti * 16 + v + 8 * chi;
    int c0 = tj0 * 16 + m;
    int c1 = tj1 * 16 + m;
    float x0 = acc0[v] + (r == c0 ? REG : 0.f);
    float x1 = acc1[v] + (r == c1 ? REG : 0.f);
    Jt[r][c0] = x0;
    Jt[r][c1] = x1;
    grow[r * DD + c0] = x0;
    grow[r * DD + c1] = x1;
  }
  __syncthreads();
  if (tid < 64) {
    float s = 0.f;
#pragma unroll
    for (int e = 0; e < DD; ++e) s += Jt[tid][e] * zn[e];
    gz[tid] = s;
    grow[DD * DD + tid] = -2.f * s;
  }
  __syncthreads();
  if (tid == 0) {
    float a = 0.f;
#pragma unroll
    for (int d = 0; d < DD; ++d) a += zn[d] * gz[d];
    a_arr[n] = a;
  }
}

// ---------------------------------------------------------------------------
// Kernel 3: Y_ext row j: cols [0,4096) = z_j z_j^T, cols [4096,4160) = z_j
// ---------------------------------------------------------------------------
__global__ __launch_bounds__(256) void y_kernel(const float* __restrict__ z,
                                                float* __restrict__ y_ext) {
  __shared__ float zr[64];
  const int j = blockIdx.x;
  const int tid = threadIdx.x;
  if (tid < 64) zr[tid] = z[(size_t)j * DD + tid];
  __syncthreads();
  float* row = y_ext + (size_t)j * KEXT;
  for (int p = tid; p < DD * DD; p += 256) row[p] = zr[p >> 6] * zr[p & 63];
  if (tid < 64) row[DD * DD + tid] = zr[tid];
}

// ---------------------------------------------------------------------------
// Kernel 4: out[n,j] = sqrt(clamp(a[n] + Gi_ext[n,:] . Y_ext[j,:], 1e-12))
// (the dot carries zGz - 2*cross).  Block tile 64(M) x 128(N), K=4160 fp32
// WMMA; 2x2 register tile per wave (4 WMMA per 4 fragment loads).  Operands
// staged into LDS with GLOBAL_LOAD_ASYNC_TO_LDS_B128 (ASYNCcnt path).
// ---------------------------------------------------------------------------
__global__ __launch_bounds__(256) void dist_kernel(
    const float* __restrict__ gi_ext, const float* __restrict__ y_ext,
    const float* __restrict__ a_arr, float* __restrict__ out) {
  __shared__ float As[64][36];    // stride 144B -> every [r][4c] 16B aligned
  __shared__ float Bs[128][36];

  const int bx = blockIdx.x & 15;   // j tile (128 wide)
  const int by = blockIdx.x >> 4;   // i tile (64 tall)
  const int i0 = by * 64, j0 = bx * 128;
  const int tid  = threadIdx.x;
  const int lane = tid & 31;
  const int wave = tid >> 5;
  const int m    = lane & 15;
  const int kh   = (lane >> 4) << 1;
  const int chi  = lane >> 4;

  // 2x2 tiles per wave: rows {2rg,2rg+1}, cols {2cg,2cg+1} (16x16 tiles)
  const int rg  = wave >> 2;          // 0..1
  const int cg  = wave & 3;           // 0..3
  const int ti0 = 2 * rg, ti1 = ti0 + 1;
  const int tj0 = 2 * cg, tj1 = tj0 + 1;

  v8f acc00 = {}, acc01 = {}, acc10 = {}, acc11 = {};
  for (int kc = 0; kc < KEXT; kc += 32) {
    if (kc + 32 < KEXT) {  // hint next chunk toward L2 (global_prefetch_b8)
      int r = tid & 63;
      __builtin_prefetch(&gi_ext[(size_t)(i0 + r) * KEXT + kc + 32], 0, 0);
      __builtin_prefetch(&y_ext[(size_t)(j0 + (tid & 127)) * KEXT + kc + 32],
                         0, 0);
    }
    __syncthreads();
    // stage A (64x32) and B (128x32) via async B128 copies: 6 per thread
    for (int p = tid; p < 512 + 1024; p += 256) {
      if (p < 512) {
        int r = p >> 3, c = (p & 7) * 4;
        async_copy_b128(&As[r][c],
                        &gi_ext[(size_t)(i0 + r) * KEXT + kc + c]);
      } else {
        int q = p - 512;
        int r = q >> 3, c = (q & 7) * 4;
        async_copy_b128(&Bs[r][c],
                        &y_ext[(size_t)(j0 + r) * KEXT + kc + c]);
      }
    }
    async_wait0();
    __syncthreads();
#pragma unroll
    for (int kk = 0; kk < 32; kk += 4) {
      v2f a0, a1, b0, b1;
      a0.x = As[ti0 * 16 + m][kk + kh];
      a0.y = As[ti0 * 16 + m][kk + kh + 1];
      a1.x = As[ti1 * 16 + m][kk + kh];
      a1.y = As[ti1 * 16 + m][kk + kh + 1];
      b0.x = Bs[tj0 * 16 + m][kk + kh];
      b0.y = Bs[tj0 * 16 + m][kk + kh + 1];
      b1.x = Bs[tj1 * 16 + m][kk + kh];
      b1.y = Bs[tj1 * 16 + m][kk + kh + 1];
      acc00 = wmma_f32(a0, b0, acc00);
      acc01 = wmma_f32(a0, b1, acc01);
      acc10 = wmma_f32(a1, b0, acc10);
      acc11 = wmma_f32(a1, b1, acc11);
    }
  }

#pragma unroll
  for (int v = 0; v < 8; ++v) {
    int gr0 = i0 + ti0 * 16 + v + 8 * chi;
    int gr1 = i0 + ti1 * 16 + v + 8 * chi;
    float an0 = a_arr[gr0];
    float an1 = a_arr[gr1];
    int gc0 = j0 + tj0 * 16 + m;
    int gc1 = j0 + tj1 * 16 + m;
    out[(size_t)gr0 * NN + gc0] = sqrtf(fmaxf(an0 + acc00[v], 1e-12f));
    out[(size_t)gr0 * NN + gc1] = sqrtf(fmaxf(an0 + acc01[v], 1e-12f));
    out[(size_t)gr1 * NN + gc0] = sqrtf(fmaxf(an1 + acc10[v], 1e-12f));
    out[(size_t)gr1 * NN + gc1] = sqrtf(fmaxf(an1 + acc11[v], 1e-12f));
  }
}

// ---------------------------------------------------------------------------
extern "C" void kernel_launch(void* const* d_in, const int* in_sizes, int n_in,
                              void* d_out, int out_size, void* d_ws,
                              size_t ws_size, hipStream_t stream) {
  (void)in_sizes; (void)n_in; (void)out_size; (void)ws_size;
  const float* z    = (const float*)d_in[0];
  const float* grid = (const float*)d_in[1];
  const float* jac  = (const float*)d_in[2];
  float* out = (float*)d_out;

  char* ws = (char*)d_ws;
  float* gsq    = (float*)(ws);                               // 64 KB
  int*   nidx   = (int*)(ws + (64 << 10));                    // 32 KB
  float* ndist  = (float*)(ws + (96 << 10));                  // 32 KB
  float* a_arr  = (float*)(ws + (128 << 10));                 //  8 KB
  float* gi_ext = (float*)(ws + (1 << 20));                   // 34.1 MB
  float* y_ext  = (float*)(ws + (1 << 20) +
                           (size_t)NN * KEXT * sizeof(float)); // 34.1 MB

  gsq_kernel <<<GG / 256, 256, 0, stream>>>(grid, gsq);
  knn_kernel <<<NN / 16, 256, 0, stream>>>(z, grid, gsq, nidx, ndist);
  gi_kernel  <<<NN, 256, 0, stream>>>(z, jac, nidx, ndist, gi_ext, a_arr);
  y_kernel   <<<NN, 256, 0, stream>>>(z, y_ext);
  dist_kernel<<<(NN / 64) * (NN / 128), 256, 0, stream>>>(gi_ext, y_ext,
                                                          a_arr, out);
}